// GCNemb_17291538334379
// MI455X (gfx1250) — compile-verified
//
#include <hip/hip_runtime.h>
#include <math.h>

#define NNODES 100000
#define NEDGES 1600000
#define FDIM   64

typedef __attribute__((ext_vector_type(2))) float v2f;
typedef __attribute__((ext_vector_type(8))) float v8f;

// ---------------- degree / norm ----------------

__global__ void zero_kernel(float* __restrict__ p, int n) {
  int i = blockIdx.x * blockDim.x + threadIdx.x;
  if (i < n) p[i] = 0.0f;
}

__global__ void deg_kernel(const int* __restrict__ col, float* __restrict__ deg, int e) {
  int i = blockIdx.x * blockDim.x + threadIdx.x;
  if (i < e) atomicAdd(&deg[col[i]], 1.0f);
}

// dinv = rsqrt(deg + 1)  (the +1 is the self loop; always > 0)
__global__ void dinv_kernel(float* __restrict__ d, int n) {
  int i = blockIdx.x * blockDim.x + threadIdx.x;
  if (i < n) d[i] = rsqrtf(d[i] + 1.0f);
}

// ---------------- WMMA GEMM: H = elu?(X) @ W  ----------------
// X: [n,64] f32 row-major, W: [64,64] f32 row-major, Hout: [n,64]
// One wave32 computes a 16x64 tile using V_WMMA_F32_16X16X4_F32.
// A 16x4 layout : lanes 0-15 -> M=lane, v.x=K+0, v.y=K+1 ; lanes 16-31 -> M=lane-16, v.x=K+2, v.y=K+3
// B 4x16 layout : lanes 0-15 -> N=lane, v.x=K+0 row, v.y=K+1 row ; lanes 16-31 -> N=lane-16, K+2/K+3
// C 16x16 layout: VGPR r -> lanes 0-15: (M=r, N=lane) ; lanes 16-31: (M=r+8, N=lane-16)
__device__ __forceinline__ float elu1(float v) {
  return v > 0.0f ? v : expm1f(v);
}

__global__ void gemm_wmma_kernel(const float* __restrict__ X,
                                 const float* __restrict__ W,
                                 float* __restrict__ Hout,
                                 int n, int apply_elu) {
  int wave = (int)((blockIdx.x * blockDim.x + threadIdx.x) >> 5);
  int lane = threadIdx.x & 31;
  int row0 = wave * 16;
  if (row0 >= n) return;                 // uniform per wave: EXEC stays all-1s

  int mlo = lane & 15;                   // A row / B,C column within tile
  int hi  = lane >> 4;                   // 0: K+0/K+1 (M=r), 1: K+2/K+3 (M=r+8)

  v8f c0 = {}; v8f c1 = {}; v8f c2 = {}; v8f c3 = {};
  const float* arow = X + (size_t)(row0 + mlo) * FDIM;

#pragma unroll
  for (int k = 0; k < FDIM; k += 4) {
    int ka = k + 2 * hi;
    v2f a;
    a.x = arow[ka];
    a.y = arow[ka + 1];
    if (apply_elu) { a.x = elu1(a.x); a.y = elu1(a.y); }

    const float* w0 = W + (size_t)ka * FDIM;          // row K
    const float* w1 = W + (size_t)(ka + 1) * FDIM;    // row K+1
    v2f b0, b1, b2, b3;
    b0.x = w0[mlo +  0]; b0.y = w1[mlo +  0];
    b1.x = w0[mlo + 16]; b1.y = w1[mlo + 16];
    b2.x = w0[mlo + 32]; b2.y = w1[mlo + 32];
    b3.x = w0[mlo + 48]; b3.y = w1[mlo + 48];

    c0 = __builtin_amdgcn_wmma_f32_16x16x4_f32(false, a, false, b0, (short)0, c0, false, false);
    c1 = __builtin_amdgcn_wmma_f32_16x16x4_f32(false, a, false, b1, (short)0, c1, false, false);
    c2 = __builtin_amdgcn_wmma_f32_16x16x4_f32(false, a, false, b2, (short)0, c2, false, false);
    c3 = __builtin_amdgcn_wmma_f32_16x16x4_f32(false, a, false, b3, (short)0, c3, false, false);
  }

  int rbase = row0 + (hi << 3);
#pragma unroll
  for (int r = 0; r < 8; ++r) {
    float* dst = Hout + (size_t)(rbase + r) * FDIM + mlo;
    dst[0]  = c0[r];
    dst[16] = c1[r];
    dst[32] = c2[r];
    dst[48] = c3[r];
  }
}

// ---------------- out = b + h * dinv^2 (bias + self-loop term) ----------------

__global__ void init_out_kernel(const float* __restrict__ h,
                                const float* __restrict__ dinv,
                                const float* __restrict__ bias,
                                float* __restrict__ out, int n) {
  int i = blockIdx.x * blockDim.x + threadIdx.x;
  if (i >= n * FDIM) return;
  int node = i >> 6;
  int f    = i & 63;
  float di = dinv[node];
  out[i] = bias[f] + h[i] * di * di;
}

// ---------------- edge scatter: out[col] += h[row] * dinv[row]*dinv[col] ----------------
// 32 lanes per edge, 2 floats per lane (float2 gather, 2x global_atomic_add_f32)

__global__ void scatter_kernel(const float* __restrict__ h,
                               const int* __restrict__ row,
                               const int* __restrict__ col,
                               const float* __restrict__ dinv,
                               float* __restrict__ out, int e) {
  long long tid = (long long)blockIdx.x * blockDim.x + threadIdx.x;
  long long edge = tid >> 5;
  if (edge >= e) return;
  int f2 = (int)(tid & 31) * 2;
  int r = row[edge];
  int c = col[edge];
  float nm = dinv[r] * dinv[c];
  const float2 v = *(const float2*)(h + (size_t)r * FDIM + f2);
  float* dst = out + (size_t)c * FDIM + f2;
  atomicAdd(dst + 0, v.x * nm);
  atomicAdd(dst + 1, v.y * nm);
}

// ---------------- driver ----------------

extern "C" void kernel_launch(void* const* d_in, const int* in_sizes, int n_in,
                              void* d_out, int out_size, void* d_ws, size_t ws_size,
                              hipStream_t stream) {
  const float* x   = (const float*)d_in[0];
  const int*   ei  = (const int*)d_in[1];       // [2, E] row-major
  const int*   row = ei;                         // sources
  const int*   col = ei + NEDGES;                // targets
  const float* W1  = (const float*)d_in[2];
  const float* b1  = (const float*)d_in[3];
  const float* W2  = (const float*)d_in[4];
  const float* b2  = (const float*)d_in[5];
  const float* W3  = (const float*)d_in[6];
  const float* b3  = (const float*)d_in[7];
  float* out = (float*)d_out;

  float* dinv = (float*)d_ws;                       // N
  float* hbuf = dinv + NNODES;                      // N*64
  float* a1   = hbuf + (size_t)NNODES * FDIM;       // N*64
  float* a2   = a1   + (size_t)NNODES * FDIM;       // N*64

  const int TB = 256;
  int nblk_n  = (NNODES + TB - 1) / TB;
  int nblk_e  = (NEDGES + TB - 1) / TB;
  int nblk_nf = (NNODES * FDIM + TB - 1) / TB;
  int nblk_sc = (int)(((long long)NEDGES * 32 + TB - 1) / TB);
  int nblk_gm = (NNODES / 16 + 7) / 8;              // 8 waves per block

  // normalization
  zero_kernel<<<nblk_n, TB, 0, stream>>>(dinv, NNODES);
  deg_kernel <<<nblk_e, TB, 0, stream>>>(col, dinv, NEDGES);
  dinv_kernel<<<nblk_n, TB, 0, stream>>>(dinv, NNODES);

  // layer 1
  gemm_wmma_kernel<<<nblk_gm, TB, 0, stream>>>(x, W1, hbuf, NNODES, 0);
  init_out_kernel <<<nblk_nf, TB, 0, stream>>>(hbuf, dinv, b1, a1, NNODES);
  scatter_kernel  <<<nblk_sc, TB, 0, stream>>>(hbuf, row, col, dinv, a1, NEDGES);

  // layer 2 (ELU fused into GEMM A-load)
  gemm_wmma_kernel<<<nblk_gm, TB, 0, stream>>>(a1, W2, hbuf, NNODES, 1);
  init_out_kernel <<<nblk_nf, TB, 0, stream>>>(hbuf, dinv, b2, a2, NNODES);
  scatter_kernel  <<<nblk_sc, TB, 0, stream>>>(hbuf, row, col, dinv, a2, NEDGES);

  // layer 3
  gemm_wmma_kernel<<<nblk_gm, TB, 0, stream>>>(a2, W3, hbuf, NNODES, 1);
  init_out_kernel <<<nblk_nf, TB, 0, stream>>>(hbuf, dinv, b3, out, NNODES);
  scatter_kernel  <<<nblk_sc, TB, 0, stream>>>(hbuf, row, col, dinv, out, NEDGES);
}